// Edge_aggregation_layer_49512382988573
// MI455X (gfx1250) — compile-verified
//
#include <hip/hip_runtime.h>
#include <hip/hip_bf16.h>

typedef __attribute__((ext_vector_type(2))) float v2f;
typedef __attribute__((ext_vector_type(8))) float v8f;

#define EDGES_PER_BLOCK 32   // two 16-row M tiles per wave -> B-fragment reuse x2
#define NODE_DIM 128
#define EDGE_ATTR_DIM 64
#define IN_DIM 320           // 2*NODE_DIM + EDGE_ATTR_DIM
#define OUT_DIM 128
#define APITCH 324           // IN_DIM + 4 : stride mod 64 == 4 -> conflict-free ds_load_b64
#define HPITCH 132           // OUT_DIM + 4

__global__ __launch_bounds__(256)
void edge_mlp_wmma_f32(const float*  __restrict__ x,
                       const long long* __restrict__ edge_index,
                       const float*  __restrict__ edge_attr,
                       const float*  __restrict__ W1,
                       const float*  __restrict__ b1,
                       const float*  __restrict__ W2,
                       const float*  __restrict__ b2,
                       float*        __restrict__ out,
                       int n_edges)
{
    __shared__ float A[EDGES_PER_BLOCK * APITCH];   // gathered edge inputs [32 x 320]
    __shared__ float H[EDGES_PER_BLOCK * HPITCH];   // hidden activations  [32 x 128]
    __shared__ int   s_src[EDGES_PER_BLOCK];
    __shared__ int   s_tgt[EDGES_PER_BLOCK];

    const int tid = threadIdx.x;
    const int e0  = blockIdx.x * EDGES_PER_BLOCK;

    // ---- load edge indices for this tile (int64 -> int) ----
    if (tid < EDGES_PER_BLOCK) {
        int eg = e0 + tid;
        if (eg >= n_edges) eg = n_edges - 1;              // clamp (tail-safe)
        s_src[tid] = (int)edge_index[eg];                 // row 0: src
        s_tgt[tid] = (int)edge_index[(size_t)n_edges + eg]; // row 1: tgt
    }
    __syncthreads();

    // ---- cooperative gather: 32 edges x 80 float4 chunks = 2560 chunks ----
    #pragma unroll
    for (int it = 0; it < 10; ++it) {
        int c = tid + it * 256;                   // 10*256 = 2560 exactly
        int e = c / 80;
        int f = c % 80;                           // float4 index within 320-float row
        int eg = e0 + e; if (eg >= n_edges) eg = n_edges - 1;
        const float4* sp;
        if (f < 32)
            sp = (const float4*)(x + (size_t)s_src[e] * NODE_DIM) + f;
        else if (f < 64)
            sp = (const float4*)(x + (size_t)s_tgt[e] * NODE_DIM) + (f - 32);
        else
            sp = (const float4*)(edge_attr + (size_t)eg * EDGE_ATTR_DIM) + (f - 64);
        *(float4*)(&A[e * APITCH + f * 4]) = *sp;
    }
    __syncthreads();

    // ---- per-wave tile assignment ----
    const int wave = tid >> 5;            // 0..7 : which 16-wide slab of the 128 outputs
    const int lane = tid & 31;
    const int m    = lane & 15;           // A: row-within-tile,  B/D: column-within-slab
    const int hi   = lane >> 4;           // 0/1
    const int koff = hi << 1;             // K sub-offset 0 or 2 (ISA 16x4 layout)
    const int ncol = wave * 16 + m;       // global output column of this lane

    // ===== GEMM1: h = relu(A[32x320] @ W1[320x128] + b1), two M tiles/wave =====
    v8f acc0 = {};
    v8f acc1 = {};
    {
        const float* a0 = &A[ m       * APITCH + koff];  // edges 0..15
        const float* a1 = &A[(m + 16) * APITCH + koff];  // edges 16..31
        const float* wb = W1 + ncol;                     // W1[k][ncol] = W1[k*128 + ncol]
        #pragma unroll 4
        for (int k = 0; k < IN_DIM; k += 4) {
            v2f bf  = { wb[(size_t)(k + koff)     * OUT_DIM],
                        wb[(size_t)(k + koff + 1) * OUT_DIM] };
            v2f af0 = { a0[k], a0[k + 1] };
            v2f af1 = { a1[k], a1[k + 1] };
            acc0 = __builtin_amdgcn_wmma_f32_16x16x4_f32(
                       false, af0, false, bf, (short)0, acc0, false, false);
            acc1 = __builtin_amdgcn_wmma_f32_16x16x4_f32(
                       false, af1, false, bf, (short)0, acc1, false, false);
        }
    }
    // bias + ReLU, write hidden tiles per C/D layout: row = r + 8*hi (+16), col = ncol
    {
        const float bias = b1[ncol];
        #pragma unroll
        for (int r = 0; r < 8; ++r) {
            float v0 = acc0[r] + bias;
            float v1 = acc1[r] + bias;
            H[(r + 8 * hi)      * HPITCH + ncol] = v0 > 0.0f ? v0 : 0.0f;
            H[(r + 8 * hi + 16) * HPITCH + ncol] = v1 > 0.0f ? v1 : 0.0f;
        }
    }
    __syncthreads();

    // ===== GEMM2: out = H[32x128] @ W2[128x128] + b2, two M tiles/wave =====
    v8f oac0 = {};
    v8f oac1 = {};
    {
        const float* h0 = &H[ m       * HPITCH + koff];
        const float* h1 = &H[(m + 16) * HPITCH + koff];
        const float* wb = W2 + ncol;                     // W2[k][ncol]
        #pragma unroll 4
        for (int k = 0; k < OUT_DIM; k += 4) {
            v2f bf  = { wb[(size_t)(k + koff)     * OUT_DIM],
                        wb[(size_t)(k + koff + 1) * OUT_DIM] };
            v2f hf0 = { h0[k], h0[k + 1] };
            v2f hf1 = { h1[k], h1[k + 1] };
            oac0 = __builtin_amdgcn_wmma_f32_16x16x4_f32(
                       false, hf0, false, bf, (short)0, oac0, false, false);
            oac1 = __builtin_amdgcn_wmma_f32_16x16x4_f32(
                       false, hf1, false, bf, (short)0, oac1, false, false);
        }
    }
    // ---- epilogue: add b2, store both D tiles ----
    {
        const float bias = b2[ncol];
        #pragma unroll
        for (int r = 0; r < 8; ++r) {
            int row0 = r + 8 * hi;                // edge within tile (0..15)
            int eg0  = e0 + row0;
            int eg1  = eg0 + 16;
            if (eg0 < n_edges)
                out[(size_t)eg0 * OUT_DIM + ncol] = oac0[r] + bias;
            if (eg1 < n_edges)
                out[(size_t)eg1 * OUT_DIM + ncol] = oac1[r] + bias;
        }
    }
}

extern "C" void kernel_launch(void* const* d_in, const int* in_sizes, int n_in,
                              void* d_out, int out_size, void* d_ws, size_t ws_size,
                              hipStream_t stream) {
    // setup_inputs() order: x, edge_index(int64), edge_attr, W1, b1, W2, b2
    const float*     x          = (const float*)d_in[0];
    const long long* edge_index = (const long long*)d_in[1];
    const float*     edge_attr  = (const float*)d_in[2];
    const float*     W1         = (const float*)d_in[3];
    const float*     b1         = (const float*)d_in[4];
    const float*     W2         = (const float*)d_in[5];
    const float*     b2         = (const float*)d_in[6];
    float*           out        = (float*)d_out;

    const int n_edges = in_sizes[2] / EDGE_ATTR_DIM;   // edge_attr is [E,64]
    const int grid    = (n_edges + EDGES_PER_BLOCK - 1) / EDGES_PER_BLOCK;

    edge_mlp_wmma_f32<<<grid, 256, 0, stream>>>(
        x, edge_index, edge_attr, W1, b1, W2, b2, out, n_edges);
}